// GraphConvolution_5815385719421
// MI455X (gfx1250) — compile-verified
//
#include <hip/hip_runtime.h>
#include <hip/hip_bf16.h>

// ---------------------------------------------------------------------------
// GCN layer on MI455X (gfx1250, wave32):
//   support = X @ W          (fp32 WMMA 16x16x4, K-loop over 128)
//   out     = bias           (broadcast init)
//   out[r] += v * support[c] (edge-parallel, native f32 atomics into L2)
// ---------------------------------------------------------------------------

typedef float v2f __attribute__((ext_vector_type(2)));
typedef float v8f __attribute__((ext_vector_type(8)));

#define D_DIM 128   // D_IN == D_OUT == 128

// ---------------------------------------------------------------------------
// Kernel 1: support = X @ W using V_WMMA_F32_16X16X4_F32.
// One wave computes one 16x16 tile of support. Block = 256 threads = 8 waves
// = the 8 column tiles (8*16 = 128) of one 16-row stripe. Grid = nRows/16.
//
// VGPR layouts per CDNA5 ISA 7.12.2 (wave32):
//   A 16x4 f32 : lane L -> row M = L%16, VGPRj holds K = 2*(L/16) + j
//   B  4x16 f32: lane L -> col N = L%16, VGPRj holds K = 2*(L/16) + j
//   C/D 16x16  : lane L -> col N = L%16, VGPRi holds M = i + 8*(L/16)
// ---------------------------------------------------------------------------
__global__ void __launch_bounds__(256, 4)
gcn_gemm_wmma_f32(const float* __restrict__ X,
                  const float* __restrict__ W,
                  float* __restrict__ S)
{
    const int lane = threadIdx.x & 31;
    const int wave = threadIdx.x >> 5;       // 0..7 -> 16-wide column tile
    const int half = lane >> 4;              // 0 or 1
    const int l16  = lane & 15;

    const int row0 = blockIdx.x << 4;        // 16-row stripe
    const int col0 = wave << 4;              // 16-col tile

    // Per-lane base pointers (pair of K elements owned by this lane).
    const float* aptr = X + (size_t)(row0 + l16) * D_DIM + (half << 1);
    const float* bptr = W + (size_t)(half << 1) * D_DIM + col0 + l16;

    v8f acc = {};
#pragma unroll 8
    for (int k = 0; k < D_DIM; k += 4) {
        v2f a, b;
        // A pair: contiguous -> single global_load_b64
        a.x = aptr[k + 0];
        a.y = aptr[k + 1];
        // B pair: two rows of W (stride D_DIM)
        b.x = bptr[(size_t)k * D_DIM + 0];
        b.y = bptr[(size_t)k * D_DIM + D_DIM];
        acc = __builtin_amdgcn_wmma_f32_16x16x4_f32(
            /*neg_a=*/false, a, /*neg_b=*/false, b,
            /*c_mod=*/(short)0, acc, /*reuse_a=*/false, /*reuse_b=*/false);
    }

    // Store D: VGPR i -> row (row0 + i + 8*half), col (col0 + l16)
    float* out = S + (size_t)(row0 + (half << 3)) * D_DIM + col0 + l16;
#pragma unroll
    for (int i = 0; i < 8; ++i)
        out[(size_t)i * D_DIM] = acc[i];
}

// ---------------------------------------------------------------------------
// Kernel 2: out[n][:] = bias[:]  (float4 vectorized; 32 chunks per row)
// ---------------------------------------------------------------------------
__global__ void __launch_bounds__(256)
gcn_bias_init(const float* __restrict__ bias,
              float* __restrict__ out,
              int nChunks)   // nNodes * 32
{
    int idx = blockIdx.x * blockDim.x + threadIdx.x;
    if (idx >= nChunks) return;
    const float4 b = ((const float4*)bias)[idx & 31];
    ((float4*)out)[idx] = b;
}

// ---------------------------------------------------------------------------
// Kernel 3: for each edge e: out[row[e]][:] += val[e] * support[col[e]][:]
// One wave32 per edge; lane owns a float4 chunk (32*4 = 128 floats).
// support fits in the 192MB L2, so the random gather hits L2; the atomics
// are non-returning global_atomic_add_f32, resolved at the L2 atomic units.
// ---------------------------------------------------------------------------
__global__ void __launch_bounds__(256)
gcn_spmm_edges(const float* __restrict__ S,
               const float* __restrict__ val,
               const int*   __restrict__ erow,
               const int*   __restrict__ ecol,
               float* __restrict__ out,
               int nEdges)
{
    const int wave = (int)((blockIdx.x * (unsigned)blockDim.x + threadIdx.x) >> 5);
    const int lane = threadIdx.x & 31;
    if (wave >= nEdges) return;

    const int   r = erow[wave];
    const int   c = ecol[wave];
    const float v = val[wave];

    const float4 m = ((const float4*)(S + (size_t)c * D_DIM))[lane];

    float* dst = out + (size_t)r * D_DIM + (lane << 2);
    unsafeAtomicAdd(dst + 0, v * m.x);
    unsafeAtomicAdd(dst + 1, v * m.y);
    unsafeAtomicAdd(dst + 2, v * m.z);
    unsafeAtomicAdd(dst + 3, v * m.w);
}

// ---------------------------------------------------------------------------
// Launch
// ---------------------------------------------------------------------------
extern "C" void kernel_launch(void* const* d_in, const int* in_sizes, int n_in,
                              void* d_out, int out_size, void* d_ws, size_t ws_size,
                              hipStream_t stream)
{
    const float* X    = (const float*)d_in[0];   // [N, 128]
    const float* W    = (const float*)d_in[1];   // [128, 128]
    const float* bias = (const float*)d_in[2];   // [128]
    const float* eval = (const float*)d_in[3];   // [E]
    const int*   erow = (const int*)d_in[4];     // [E]
    const int*   ecol = (const int*)d_in[5];     // [E]
    float*       out  = (float*)d_out;           // [N, 128]

    const int nNodes = in_sizes[0] / D_DIM;      // 50000 (multiple of 16)
    const int nEdges = in_sizes[3];              // 800000

    float* support = (float*)d_ws;               // [N, 128] scratch

    // 1) support = X @ W   (one block per 16-row stripe, 8 waves cover cols)
    gcn_gemm_wmma_f32<<<nNodes / 16, 256, 0, stream>>>(X, W, support);

    // 2) out = bias (broadcast)
    {
        const int nChunks = nNodes * (D_DIM / 4);
        gcn_bias_init<<<(nChunks + 255) / 256, 256, 0, stream>>>(bias, out, nChunks);
    }

    // 3) out[r] += v * support[c]   (one wave per edge)
    {
        const long long threads = (long long)nEdges * 32;
        const int blocks = (int)((threads + 255) / 256);
        gcn_spmm_edges<<<blocks, 256, 0, stream>>>(support, eval, erow, ecol, out, nEdges);
    }
}